// LieBatchNormSPD_5497558139263
// MI455X (gfx1250) — compile-verified
//
#include <hip/hip_runtime.h>
#include <math.h>

// ---------------------------------------------------------------------------
// LieBatchNormSPD on MI455X (gfx1250): B=16384 SPD matrices of size 32x32.
// wave32 == matrix dimension: one wave per matrix. Jacobi eigh in LDS,
// all 32x32 matmuls via V_WMMA_F32_16X16X4_F32 (2x2 tiles x 8 K-steps).
// Global traffic vectorized to b128; batch reductions use 64-way partial
// accumulators to avoid L2 same-address atomic serialization.
// ---------------------------------------------------------------------------

#define NN 32
#define LS 36            // LDS row stride in floats (conflict-free, 16B-aligned rows)
#define NSWEEP 8         // Jacobi sweeps
#define EPSV 1e-5f
#define NPART 64         // partial accumulator copies for batch reductions

// workspace float offsets
#define WS_BMACC  0                    // 1024: sum of X over batch
#define WS_BMSQ   1024                 // 1024: bm^{1/2}
#define WS_BMISQ  2048                 // 1024: bm^{-1/2}
#define WS_LINV   3072                 // 1024: chol(batch_mean)^{-1}
#define WS_LW     4096                 // 1024: chol(weight)
#define WS_FACTOR 5120                 // 1:    shift / sqrt(var + eps)
#define WS_VARP   5184                 // 64:   partial sums of dists
#define WS_GTP    8192                 // 64*1024: partial sums of XT
#define WS_ZERO_N (WS_GTP + NPART*1024)  // floats zeroed each launch (73728)
#define WS_LOGW   WS_ZERO_N            // B*32: per-batch log-eigenvalues of Xc

typedef __attribute__((ext_vector_type(2))) float v2f;
typedef __attribute__((ext_vector_type(8))) float v8f;

// ---------------------------------------------------------------------------
// 32x32 matmul helpers via f32 WMMA. A,B,C are LDS matrices (row-major, LS).
// C must not alias A or B. Fragment layouts per CDNA5 ISA 7.12.2 (f32):
//   A 16x4: lane m=lane&15; K pair = 2*(lane>>4)   (contiguous -> b64)
//   B 4x16: lane n=lane&15; K = vgpr + 2*(lane>>4)
//   C/D:    lane n=lane&15; M = r + 8*(lane>>4)
// ---------------------------------------------------------------------------
__device__ __forceinline__ void wave_mm_nn(const float* A, const float* Bm,
                                           float* C, int lane) {
  const int half = lane >> 4;  // 0|1
  const int l15 = lane & 15;
  #pragma unroll
  for (int ti = 0; ti < 2; ++ti) {
    #pragma unroll
    for (int tj = 0; tj < 2; ++tj) {
      v8f acc = {};
      const int m = l15 + 16 * ti;
      const int n = l15 + 16 * tj;
      #pragma unroll
      for (int kb = 0; kb < 8; ++kb) {
        const int ka = kb * 4 + 2 * half;
        v2f a; a.x = A[m * LS + ka];         a.y = A[m * LS + ka + 1];
        v2f b; b.x = Bm[(ka + 0) * LS + n];  b.y = Bm[(ka + 1) * LS + n];
        acc = __builtin_amdgcn_wmma_f32_16x16x4_f32(
            false, a, false, b, (short)0, acc, false, false);
      }
      #pragma unroll
      for (int r = 0; r < 8; ++r) {
        const int mr = r + 8 * half + 16 * ti;
        C[mr * LS + l15 + 16 * tj] = acc[r];
      }
    }
  }
}

// C = A * B^T
__device__ __forceinline__ void wave_mm_nt(const float* A, const float* Bm,
                                           float* C, int lane) {
  const int half = lane >> 4;
  const int l15 = lane & 15;
  #pragma unroll
  for (int ti = 0; ti < 2; ++ti) {
    #pragma unroll
    for (int tj = 0; tj < 2; ++tj) {
      v8f acc = {};
      const int m = l15 + 16 * ti;
      const int n = l15 + 16 * tj;
      #pragma unroll
      for (int kb = 0; kb < 8; ++kb) {
        const int ka = kb * 4 + 2 * half;
        v2f a; a.x = A[m * LS + ka];      a.y = A[m * LS + ka + 1];
        v2f b; b.x = Bm[n * LS + ka];     b.y = Bm[n * LS + ka + 1];  // B^T[k][n]
        acc = __builtin_amdgcn_wmma_f32_16x16x4_f32(
            false, a, false, b, (short)0, acc, false, false);
      }
      #pragma unroll
      for (int r = 0; r < 8; ++r) {
        const int mr = r + 8 * half + 16 * ti;
        C[mr * LS + l15 + 16 * tj] = acc[r];
      }
    }
  }
}

// ---------------------------------------------------------------------------
// Vectorized global<->LDS tile movers (b128). LS rows are 144B (16B aligned).
// ---------------------------------------------------------------------------
__device__ __forceinline__ void load_tile(const float* __restrict__ g,
                                          float* lds, int lane) {
  #pragma unroll
  for (int i = 0; i < 8; ++i) {
    const int linear = i * NN + lane;        // float4 index within 256
    const int row = linear >> 3;
    const int c4 = linear & 7;
    *(float4*)&lds[row * LS + c4 * 4] = *(const float4*)&g[linear * 4];
  }
}

__device__ __forceinline__ void store_tile(float* __restrict__ g,
                                           const float* lds, int lane) {
  #pragma unroll
  for (int i = 0; i < 8; ++i) {
    const int linear = i * NN + lane;
    const int row = linear >> 3;
    const int c4 = linear & 7;
    *(float4*)&g[linear * 4] = *(const float4*)&lds[row * LS + c4 * 4];
  }
}

// ---------------------------------------------------------------------------
// Cyclic Jacobi eigendecomposition of symmetric 32x32 A (destroyed),
// eigenvectors -> V, eigenvalues -> w[32]. One wave. Round-robin schedule:
// 16 disjoint pivot pairs per round, 31 rounds per sweep.
// ---------------------------------------------------------------------------
__device__ void wave_eigh(float* A, float* V, float* w,
                          float* cosA, float* sinA, int* top, int* bot,
                          int lane) {
  for (int r = 0; r < NN; ++r) V[r * LS + lane] = (r == lane) ? 1.0f : 0.0f;
  if (lane < 16) { top[lane] = 2 * lane; bot[lane] = 2 * lane + 1; }
  __syncthreads();

  for (int sweep = 0; sweep < NSWEEP; ++sweep) {
    #pragma unroll 1
    for (int round = 0; round < NN - 1; ++round) {
      // rotation angles for the 16 pairs
      if (lane < 16) {
        const int p = top[lane], q = bot[lane];
        const float app = A[p * LS + p];
        const float aqq = A[q * LS + q];
        const float apq = A[p * LS + q];
        float c = 1.0f, s = 0.0f;
        if (fabsf(apq) > 1e-20f) {
          const float tau = (aqq - app) / (2.0f * apq);
          const float tt = 1.0f / (fabsf(tau) + sqrtf(1.0f + tau * tau));
          const float t = (tau >= 0.0f) ? tt : -tt;
          c = 1.0f / sqrtf(1.0f + t * t);
          s = t * c;
        }
        cosA[lane] = c; sinA[lane] = s;
      }
      __syncthreads();
      // row rotations (disjoint rows across pairs): lane = column index
      #pragma unroll 1
      for (int i = 0; i < 16; ++i) {
        const int p = top[i], q = bot[i];
        const float c = cosA[i], s = sinA[i];
        const float ap = A[p * LS + lane], aq = A[q * LS + lane];
        A[p * LS + lane] = c * ap - s * aq;
        A[q * LS + lane] = s * ap + c * aq;
      }
      __syncthreads();
      // column rotations + eigenvector update: lane = row index
      #pragma unroll 1
      for (int i = 0; i < 16; ++i) {
        const int p = top[i], q = bot[i];
        const float c = cosA[i], s = sinA[i];
        const float ap = A[lane * LS + p], aq = A[lane * LS + q];
        A[lane * LS + p] = c * ap - s * aq;
        A[lane * LS + q] = s * ap + c * aq;
        const float vp = V[lane * LS + p], vq = V[lane * LS + q];
        V[lane * LS + p] = c * vp - s * vq;
        V[lane * LS + q] = s * vp + c * vq;
      }
      __syncthreads();
      // rotate round-robin schedule (circle method, top[0] fixed)
      int newt = 0, newb = 0;
      if (lane < 16) {
        newt = (lane == 0) ? top[0] : ((lane == 1) ? bot[0] : top[lane - 1]);
        newb = (lane == 15) ? top[15] : bot[lane + 1];
      }
      __syncthreads();
      if (lane < 16) { top[lane] = newt; bot[lane] = newb; }
      __syncthreads();
    }
  }
  w[lane] = A[lane * LS + lane];
  __syncthreads();
}

// Cholesky of SPD M -> lower L (both LDS, stride LS). lane = row owner.
__device__ void wave_chol(const float* M, float* L, int lane) {
  for (int c = 0; c < NN; ++c) L[lane * LS + c] = 0.0f;
  __syncthreads();
  for (int k = 0; k < NN; ++k) {
    if (lane == k) {
      float s = M[k * LS + k];
      for (int m = 0; m < k; ++m) { const float v = L[k * LS + m]; s -= v * v; }
      L[k * LS + k] = sqrtf(fmaxf(s, 1e-20f));
    }
    __syncthreads();
    if (lane > k) {
      float s = M[lane * LS + k];
      for (int m = 0; m < k; ++m) s -= L[lane * LS + m] * L[k * LS + m];
      L[lane * LS + k] = s / L[k * LS + k];
    }
    __syncthreads();
  }
}

// Minv = L^{-1} for lower-triangular L. lane = column owner (independent solves).
__device__ void wave_trinv(const float* L, float* Minv, int lane) {
  const int c = lane;
  for (int j = 0; j < NN; ++j) {
    float s = (j == c) ? 1.0f : 0.0f;
    for (int m = c; m < j; ++m) s -= L[j * LS + m] * Minv[m * LS + c];
    Minv[j * LS + c] = (j >= c) ? (s / L[j * LS + j]) : 0.0f;
  }
  __syncthreads();
}

// ---------------------------------------------------------------------------
// Kernels
// ---------------------------------------------------------------------------
__global__ void k_init(float* ws) {
  const int i = blockIdx.x * blockDim.x + threadIdx.x;
  if (i < WS_ZERO_N) ws[i] = 0.0f;
}

__global__ void k_bmsum(const float* __restrict__ X, float* ws, int Bt) {
  const int tid = threadIdx.x;                    // 256 threads: one float4 each
  const int per = (Bt + gridDim.x - 1) / gridDim.x;
  const int b0 = blockIdx.x * per;
  int b1 = b0 + per; if (b1 > Bt) b1 = Bt;
  float a0 = 0.f, a1 = 0.f, a2 = 0.f, a3 = 0.f;
  for (int b = b0; b < b1; ++b) {
    const float4 v = *(const float4*)&X[(size_t)b * (NN * NN) + tid * 4];
    a0 += v.x; a1 += v.y; a2 += v.z; a3 += v.w;
  }
  atomicAdd(&ws[WS_BMACC + tid * 4 + 0], a0);
  atomicAdd(&ws[WS_BMACC + tid * 4 + 1], a1);
  atomicAdd(&ws[WS_BMACC + tid * 4 + 2], a2);
  atomicAdd(&ws[WS_BMACC + tid * 4 + 3], a3);
}

// eigh(bm) -> bm^{1/2}, bm^{-1/2}
__global__ __launch_bounds__(32) void k_prep1(float* ws, int Bt) {
  __shared__ float BufA[NN * LS], BufT[NN * LS], BufX[NN * LS], BufV[NN * LS];
  __shared__ float warr[NN], cosA[16], sinA[16];
  __shared__ int top[16], bot[16];
  const int lane = threadIdx.x;
  const float inv = 1.0f / (float)Bt;
  for (int r = 0; r < NN; ++r)
    BufA[r * LS + lane] = ws[WS_BMACC + r * NN + lane] * inv;
  __syncthreads();
  wave_eigh(BufA, BufV, warr, cosA, sinA, top, bot, lane);
  const float sw = sqrtf(fmaxf(warr[lane], 1e-30f));
  // bm^{1/2} = V diag(sw) V^T
  for (int r = 0; r < NN; ++r) BufT[r * LS + lane] = BufV[r * LS + lane] * sw;
  __syncthreads();
  wave_mm_nt(BufT, BufV, BufX, lane);
  __syncthreads();
  for (int r = 0; r < NN; ++r) ws[WS_BMSQ + r * NN + lane] = BufX[r * LS + lane];
  __syncthreads();
  // bm^{-1/2} = V diag(1/sw) V^T
  const float isw = 1.0f / sw;
  for (int r = 0; r < NN; ++r) BufT[r * LS + lane] = BufV[r * LS + lane] * isw;
  __syncthreads();
  wave_mm_nt(BufT, BufV, BufX, lane);
  __syncthreads();
  for (int r = 0; r < NN; ++r) ws[WS_BMISQ + r * NN + lane] = BufX[r * LS + lane];
}

// pass A: inner = bm^{-1/2} X bm^{-1/2}; XT = logm(inner); GT partial += XT
__global__ __launch_bounds__(32) void k_passA(const float* __restrict__ X,
                                              float* ws) {
  __shared__ float BufX[NN * LS], BufA[NN * LS], BufT[NN * LS], BufV[NN * LS];
  __shared__ float warr[NN], cosA[16], sinA[16];
  __shared__ int top[16], bot[16];
  const int lane = threadIdx.x;
  load_tile(X + (size_t)blockIdx.x * (NN * NN), BufX, lane);
  for (int r = 0; r < NN; ++r)
    BufA[r * LS + lane] = ws[WS_BMISQ + r * NN + lane];
  __syncthreads();
  wave_mm_nn(BufA, BufX, BufT, lane);   __syncthreads();  // bm_isq * X
  wave_mm_nn(BufT, BufA, BufX, lane);   __syncthreads();  // inner (bm_isq symmetric)
  wave_eigh(BufX, BufV, warr, cosA, sinA, top, bot, lane);
  const float lw = logf(fmaxf(warr[lane], 1e-30f));
  for (int r = 0; r < NN; ++r) BufT[r * LS + lane] = BufV[r * LS + lane] * lw;
  __syncthreads();
  wave_mm_nt(BufT, BufV, BufA, lane);   __syncthreads();  // XT = V diag(logw) V^T
  float* gt = &ws[WS_GTP + (size_t)(blockIdx.x & (NPART - 1)) * (NN * NN)];
  for (int r = 0; r < NN; ++r)
    atomicAdd(&gt[r * NN + lane], BufA[r * LS + lane]);
}

// expm(GT/B); batch_mean = bm_sq E bm_sq; Linv = chol(batch_mean)^{-1}; Lw = chol(weight)
__global__ __launch_bounds__(32) void k_central(const float* __restrict__ weight,
                                                float* ws, int Bt) {
  __shared__ float BufA[NN * LS], BufT[NN * LS], BufX[NN * LS], BufV[NN * LS];
  __shared__ float warr[NN], cosA[16], sinA[16];
  __shared__ int top[16], bot[16];
  const int lane = threadIdx.x;
  const float inv = 1.0f / (float)Bt;
  for (int r = 0; r < NN; ++r) {
    float s = 0.0f;
    for (int p = 0; p < NPART; ++p)
      s += ws[WS_GTP + p * (NN * NN) + r * NN + lane];
    BufA[r * LS + lane] = s * inv;     // GT mean
  }
  __syncthreads();
  wave_eigh(BufA, BufV, warr, cosA, sinA, top, bot, lane);
  const float ew = expf(warr[lane]);
  for (int r = 0; r < NN; ++r) BufT[r * LS + lane] = BufV[r * LS + lane] * ew;
  __syncthreads();
  wave_mm_nt(BufT, BufV, BufX, lane);   __syncthreads();  // E = expm(GT)
  for (int r = 0; r < NN; ++r) BufA[r * LS + lane] = ws[WS_BMSQ + r * NN + lane];
  __syncthreads();
  wave_mm_nn(BufA, BufX, BufT, lane);   __syncthreads();  // bm_sq * E
  wave_mm_nn(BufT, BufA, BufX, lane);   __syncthreads();  // batch_mean
  wave_chol(BufX, BufA, lane);                             // L
  wave_trinv(BufA, BufT, lane);                            // Linv
  __syncthreads();
  for (int r = 0; r < NN; ++r) ws[WS_LINV + r * NN + lane] = BufT[r * LS + lane];
  __syncthreads();
  for (int r = 0; r < NN; ++r) BufX[r * LS + lane] = weight[r * NN + lane];
  __syncthreads();
  wave_chol(BufX, BufA, lane);                             // Lw
  __syncthreads();
  for (int r = 0; r < NN; ++r) ws[WS_LW + r * NN + lane] = BufA[r * LS + lane];
}

// pass B: Xc = Linv X Linv^T; eigh(Xc); logw -> ws; vc -> out; dists -> var partials
__global__ __launch_bounds__(32) void k_passB(const float* __restrict__ X,
                                              float* ws, float* out) {
  __shared__ float BufX[NN * LS], BufA[NN * LS], BufT[NN * LS], BufV[NN * LS];
  __shared__ float warr[NN], cosA[16], sinA[16];
  __shared__ int top[16], bot[16];
  const int lane = threadIdx.x;
  const size_t b = blockIdx.x;
  load_tile(X + b * (NN * NN), BufX, lane);
  for (int r = 0; r < NN; ++r)
    BufA[r * LS + lane] = ws[WS_LINV + r * NN + lane];
  __syncthreads();
  wave_mm_nn(BufA, BufX, BufT, lane);   __syncthreads();  // Linv * X
  wave_mm_nt(BufT, BufA, BufX, lane);   __syncthreads();  // Xc = (Linv X) Linv^T
  wave_eigh(BufX, BufV, warr, cosA, sinA, top, bot, lane);
  const float lw = logf(fmaxf(warr[lane], 1e-30f));
  ws[WS_LOGW + b * NN + lane] = lw;
  warr[lane] = lw * lw;
  __syncthreads();
  if (lane == 0) {
    float s = 0.0f;
    for (int i = 0; i < NN; ++i) s += warr[i];
    atomicAdd(&ws[WS_VARP + (blockIdx.x & (NPART - 1))], s);  // ALPHA=1, BETA=0
  }
  store_tile(out + b * (NN * NN), BufV, lane);              // stage vc in d_out
}

__global__ void k_factor(const float* __restrict__ shift, float* ws, int Bt) {
  if (threadIdx.x == 0 && blockIdx.x == 0) {
    float vs = 0.0f;
    for (int p = 0; p < NPART; ++p) vs += ws[WS_VARP + p];
    const float var = vs / (float)Bt;
    ws[WS_FACTOR] = shift[0] / sqrtf(var + EPSV);
  }
}

// pass C: Xs = vc diag(exp(factor*logw)) vc^T; Xn = Lw Xs Lw^T -> out
__global__ __launch_bounds__(32) void k_passC(float* ws, float* out) {
  __shared__ float BufX[NN * LS], BufA[NN * LS], BufT[NN * LS], BufV[NN * LS];
  const int lane = threadIdx.x;
  const size_t b = blockIdx.x;
  const float factor = ws[WS_FACTOR];
  const float e = expf(factor * ws[WS_LOGW + b * NN + lane]);
  load_tile(out + b * (NN * NN), BufV, lane);               // vc
  __syncthreads();
  for (int r = 0; r < NN; ++r) BufT[r * LS + lane] = BufV[r * LS + lane] * e;
  __syncthreads();
  wave_mm_nt(BufT, BufV, BufX, lane);   __syncthreads();    // Xs
  for (int r = 0; r < NN; ++r) BufA[r * LS + lane] = ws[WS_LW + r * NN + lane];
  __syncthreads();
  wave_mm_nn(BufA, BufX, BufT, lane);   __syncthreads();    // Lw * Xs
  wave_mm_nt(BufT, BufA, BufX, lane);   __syncthreads();    // Xn = (Lw Xs) Lw^T
  store_tile(out + b * (NN * NN), BufX, lane);
}

// ---------------------------------------------------------------------------
extern "C" void kernel_launch(void* const* d_in, const int* in_sizes, int n_in,
                              void* d_out, int out_size, void* d_ws, size_t ws_size,
                              hipStream_t stream) {
  (void)n_in; (void)out_size; (void)ws_size;
  const float* X = (const float*)d_in[0];
  const float* weight = (const float*)d_in[1];
  const float* shift = (const float*)d_in[2];
  float* out = (float*)d_out;
  float* ws = (float*)d_ws;
  const int Bt = in_sizes[0] / (NN * NN);

  k_init<<<(WS_ZERO_N + 255) / 256, 256, 0, stream>>>(ws);
  k_bmsum<<<64, 256, 0, stream>>>(X, ws, Bt);
  k_prep1<<<1, 32, 0, stream>>>(ws, Bt);
  k_passA<<<Bt, 32, 0, stream>>>(X, ws);
  k_central<<<1, 32, 0, stream>>>(weight, ws, Bt);
  k_passB<<<Bt, 32, 0, stream>>>(X, ws, out);
  k_factor<<<1, 32, 0, stream>>>(shift, ws, Bt);
  k_passC<<<Bt, 32, 0, stream>>>(ws, out);
}